// Layer_71897752535430
// MI455X (gfx1250) — compile-verified
//
#include <hip/hip_runtime.h>
#include <hip/hip_bf16.h>
#include <stdint.h>

typedef float v2f __attribute__((ext_vector_type(2)));
typedef float v8f __attribute__((ext_vector_type(8)));

#define CC 4
#define MM 8192
#define NN 8192
#define KCH 32          // k elements staged per tile
#define LROW 36         // padded LDS row (floats): conflict-free b64 reads, 16B-aligned rows
#define WAVES 8         // waves per workgroup
#define STAGES (CC * (MM / KCH / WAVES))   // 4 * 32 = 128 stages per wave

// ---------------------------------------------------------------------------
// Kernel 1: batched GEMV via V_WMMA_F32_16X16X4_F32, W streamed with
// GLOBAL_LOAD_ASYNC_TO_LDS_B128 (ASYNCcnt-pipelined, double buffered,
// wave-private LDS -> no barriers on the hot path).
// Each workgroup computes forward[n0..n0+15]; its 8 waves split the K dim.
// ---------------------------------------------------------------------------
__global__ __launch_bounds__(256) void gemv_wmma_f32(
    const float* __restrict__ acts, const float* __restrict__ W,
    const float* __restrict__ bwd, const float* __restrict__ ext,
    float* __restrict__ ws_inp, float* __restrict__ ws_part) {
  __shared__ float lds[WAVES * 2 * 16 * LROW];
  __shared__ float red[WAVES][16];

  const int tid  = threadIdx.x;
  const int lane = tid & 31;
  const int wv   = __builtin_amdgcn_readfirstlane(tid >> 5);  // wave id (scalar)
  const int n0   = blockIdx.x * 16;

  float* buf0 = &lds[(wv * 2 + 0) * 16 * LROW];
  float* buf1 = &lds[(wv * 2 + 1) * 16 * LROW];

  const int rhalf = lane >> 4;        // 0 for lanes 0-15, 1 for 16-31
  const int l16   = lane & 15;
  const int grow  = lane >> 3;        // row group for fetch (4 rows / instr)
  const int gmo   = (lane & 7) * 4;   // float offset within a 32-float row seg
  const bool isl0  = (lane == 0);
  const bool isl16 = (lane == 16);

  // Low 32 bits of a flat LDS pointer == workgroup-relative LDS byte address
  // (ISA: flat aperture maps LDS via addr[31:0]).
  const uint32_t lb0 = (uint32_t)(size_t)buf0 + (uint32_t)(grow * LROW + gmo) * 4u;
  const uint32_t lb1 = (uint32_t)(size_t)buf1 + (uint32_t)(grow * LROW + gmo) * 4u;

  v8f acc = {0.f, 0.f, 0.f, 0.f, 0.f, 0.f, 0.f, 0.f};

  // ---- prologue: async-stage tile 0 (c=0, t=0 -> k0 = wv*KCH) into buf0 ----
  {
    const int k0 = wv * KCH;
    const float* ga = W + ((size_t)n0 + grow) * (size_t)MM + k0 + gmo;
#pragma unroll
    for (int i = 0; i < 4; ++i) {
      const float* gai = ga + (size_t)(4 * i) * MM;
      const uint32_t lai = lb0 + (uint32_t)(4 * i * LROW) * 4u;
      asm volatile("global_load_async_to_lds_b128 %0, %1, off"
                   :: "v"(lai), "v"(gai) : "memory");
    }
  }

  for (int s = 0; s < STAGES; ++s) {
    float* cur = (s & 1) ? buf1 : buf0;

    // ---- issue async loads for stage s+1 into the other buffer ----
    if (s + 1 < STAGES) {
      const int c  = (s + 1) >> 5;
      const int t  = (s + 1) & 31;
      const int k0 = (wv + t * WAVES) * KCH;
      const float* ga =
          W + ((size_t)c * NN + n0 + grow) * (size_t)MM + k0 + gmo;
      const uint32_t lbn = (s & 1) ? lb0 : lb1;
#pragma unroll
      for (int i = 0; i < 4; ++i) {
        const float* gai = ga + (size_t)(4 * i) * MM;
        const uint32_t lai = lbn + (uint32_t)(4 * i * LROW) * 4u;
        asm volatile("global_load_async_to_lds_b128 %0, %1, off"
                     :: "v"(lai), "v"(gai) : "memory");
      }
      // 4 newest (stage s+1) may be outstanding; stage s is complete once
      // ASYNCcnt <= 4 (async loads retire in order).
      asm volatile("s_wait_asynccnt 0x4" ::: "memory");
    } else {
      asm volatile("s_wait_asynccnt 0x0" ::: "memory");
    }

    // ---- consume stage s from LDS: 8 x wmma_f32_16x16x4_f32 ----
    {
      const int c  = s >> 5;
      const int t  = s & 31;
      const int k0 = (wv + t * WAVES) * KCH;
      const float* ap = acts + c * MM + k0;  // wave-uniform -> scalar loads
#pragma unroll
      for (int i = 0; i < 8; ++i) {
        const int kk = i * 4;
        // B fragment (4x16 tile of W): lane<16 -> {kk, kk+1} of row n0+l16,
        //                              lane>=16 -> {kk+2, kk+3}
        v2f b = *(const v2f*)&cur[l16 * LROW + kk + rhalf * 2];
        // A fragment (16x4): row 0 = acts chunk, rows 1-15 zero
        float a0 = ap[kk + 0], a1 = ap[kk + 1], a2 = ap[kk + 2], a3 = ap[kk + 3];
        v2f a;
        a.x = isl0 ? a0 : (isl16 ? a2 : 0.0f);
        a.y = isl0 ? a1 : (isl16 ? a3 : 0.0f);
        acc = __builtin_amdgcn_wmma_f32_16x16x4_f32(
            false, a, false, b, (short)0, acc, false, false);
      }
    }
  }

  // D row 0 (VGPR0, lanes 0-15) holds this wave's 16 partial outputs.
  float r = acc[0];
  if (lane < 16) red[wv][l16] = r;
  __syncthreads();

  if (wv == 0) {
    float v = 0.0f;
    if (lane < 16) {
      float ssum = 0.0f;
#pragma unroll
      for (int w = 0; w < WAVES; ++w) ssum += red[w][l16];
      const int n = n0 + l16;
      v = ssum + bwd[n] + ext[n];
      ws_inp[n] = v;
    }
    // deterministic in-wave reduction for the block's contribution to mean
#pragma unroll
    for (int off = 16; off > 0; off >>= 1) v += __shfl_xor(v, off, 32);
    if (lane == 0) ws_part[blockIdx.x] = v;
  }
}

// ---------------------------------------------------------------------------
// Kernel 2: deterministic tree-reduce of the 512 per-block partials.
// ---------------------------------------------------------------------------
__global__ __launch_bounds__(512) void reduce512(const float* __restrict__ part,
                                                 float* __restrict__ sum_out) {
  __shared__ float s[512];
  const int t = threadIdx.x;
  s[t] = part[t];
  __syncthreads();
  for (int off = 256; off > 0; off >>= 1) {
    if (t < off) s[t] += s[t + off];
    __syncthreads();
  }
  if (t == 0) sum_out[0] = s[0];
}

// ---------------------------------------------------------------------------
// Kernel 3: pointwise epilogue  tanh(max(inp - max(mean,0), 0) * 8)
// ---------------------------------------------------------------------------
__global__ __launch_bounds__(256) void epilogue(const float* __restrict__ ws_inp,
                                                const float* __restrict__ sum_in,
                                                float* __restrict__ out) {
  const int n = blockIdx.x * 256 + threadIdx.x;
  const float inh = fmaxf(sum_in[0] * (1.0f / (float)NN), 0.0f);
  const float v = ws_inp[n] - inh;
  out[n] = tanhf(fmaxf(v, 0.0f) * 8.0f);
}

extern "C" void kernel_launch(void* const* d_in, const int* in_sizes, int n_in,
                              void* d_out, int out_size, void* d_ws, size_t ws_size,
                              hipStream_t stream) {
  (void)in_sizes; (void)n_in; (void)out_size; (void)ws_size;
  const float* acts = (const float*)d_in[0];  // (C, M)
  const float* W    = (const float*)d_in[1];  // (C, N, M)
  const float* bwd  = (const float*)d_in[2];  // (N,)
  const float* ext  = (const float*)d_in[3];  // (N,)
  float* out = (float*)d_out;                 // (N,)

  float* wsf     = (float*)d_ws;
  float* ws_inp  = wsf;               // N floats
  float* ws_part = wsf + NN;          // 512 floats (fully overwritten each call)
  float* ws_sum  = wsf + NN + 512;    // 1 float   (fully overwritten each call)

  gemv_wmma_f32<<<NN / 16, 256, 0, stream>>>(acts, W, bwd, ext, ws_inp, ws_part);
  reduce512<<<1, 512, 0, stream>>>(ws_part, ws_sum);
  epilogue<<<NN / 256, 256, 0, stream>>>(ws_inp, ws_sum, out);
}